// MatchLayer_34342558499030
// MI455X (gfx1250) — compile-verified
//
#include <hip/hip_runtime.h>
#include <stdint.h>

typedef float v2f __attribute__((ext_vector_type(2)));
typedef float v4f __attribute__((ext_vector_type(4)));
typedef float v8f __attribute__((ext_vector_type(8)));

constexpr int cB = 16, cL = 1024, cD = 256, cU = 64;

// ---- dot kernel tiling ----
constexpr int KC      = 64;            // K chunk staged in LDS
constexpr int LSTR    = KC + 4;        // padded LDS row stride (dwords): conflict-free
constexpr int TILE_DW = 128 * LSTR;    // one 128-row tensor tile (dwords)
constexpr int BUF_DW  = 2 * TILE_DW;   // x1 tile + x2 tile

// D = A(16x4,f32) * B(4x16,f32) + C(16x16,f32)
#define WMMA_F32(a_, b_, c_) \
  __builtin_amdgcn_wmma_f32_16x16x4_f32(false, (a_), false, (b_), (short)0, (c_), false, false)

__device__ __forceinline__ v4f vabs4(v4f v) {
  v4f r;
  r.x = __builtin_fabsf(v.x);
  r.y = __builtin_fabsf(v.y);
  r.z = __builtin_fabsf(v.z);
  r.w = __builtin_fabsf(v.w);
  return r;
}

// -------- fused: match3 features + reciprocal cosine norms --------
// wave-per-row; rn = 1/sqrt(max(||x||^2, 1e-6)); sub_norm eps = 1e-12
__global__ __launch_bounds__(256) void match3_norms_kernel(
    const float* __restrict__ x1, const float* __restrict__ x2,
    float* __restrict__ out3, float* __restrict__ rn1, float* __restrict__ rn2) {
  const int lane = threadIdx.x & 31;
  const int row  = (blockIdx.x * 256 + threadIdx.x) >> 5;   // exact: B*L waves
  const float* p1 = x1 + (size_t)row * cD;
  const float* p2 = x2 + (size_t)row * cD;
  const int d0 = 4 * lane, d1 = 128 + 4 * lane;
  v4f a0 = *(const v4f*)(p1 + d0);
  v4f a1 = *(const v4f*)(p1 + d1);
  v4f b0 = *(const v4f*)(p2 + d0);
  v4f b1 = *(const v4f*)(p2 + d1);
  v4f s0 = a0 - b0, s1 = a1 - b1;
  float q1 = a0.x * a0.x + a0.y * a0.y + a0.z * a0.z + a0.w * a0.w +
             a1.x * a1.x + a1.y * a1.y + a1.z * a1.z + a1.w * a1.w;
  float q2 = b0.x * b0.x + b0.y * b0.y + b0.z * b0.z + b0.w * b0.w +
             b1.x * b1.x + b1.y * b1.y + b1.z * b1.z + b1.w * b1.w;
  float ss = s0.x * s0.x + s0.y * s0.y + s0.z * s0.z + s0.w * s0.w +
             s1.x * s1.x + s1.y * s1.y + s1.z * s1.z + s1.w * s1.w;
#pragma unroll
  for (int off = 16; off > 0; off >>= 1) {
    q1 += __shfl_xor(q1, off, 32);
    q2 += __shfl_xor(q2, off, 32);
    ss += __shfl_xor(ss, off, 32);
  }
  if (lane == 0) {
    rn1[row] = rsqrtf(fmaxf(q1, 1e-6f));
    rn2[row] = rsqrtf(fmaxf(q2, 1e-6f));
  }
  const float rs = rsqrtf(fmaxf(ss, 1e-12f));
  float* o = out3 + (size_t)row * (6 * cD);
  *(v4f*)(o + 0 * cD + d0) = a0;            *(v4f*)(o + 0 * cD + d1) = a1;
  *(v4f*)(o + 1 * cD + d0) = b0;            *(v4f*)(o + 1 * cD + d1) = b1;
  *(v4f*)(o + 2 * cD + d0) = a0 + b0;       *(v4f*)(o + 2 * cD + d1) = a1 + b1;
  *(v4f*)(o + 3 * cD + d0) = a0 * b0;       *(v4f*)(o + 3 * cD + d1) = a1 * b1;
  *(v4f*)(o + 4 * cD + d0) = vabs4(s0);     *(v4f*)(o + 4 * cD + d1) = vabs4(s1);
  *(v4f*)(o + 5 * cD + d0) = s0 * rs;       *(v4f*)(o + 5 * cD + d1) = s1 * rs;
}

// -------- T[b] = x2[b]^T @ w1 : (D x U), K = L --------
__global__ __launch_bounds__(256) void xtw_kernel(
    const float* __restrict__ x2, const float* __restrict__ w1,
    float* __restrict__ T) {
  const int lane = threadIdx.x & 31;
  const int wv   = (blockIdx.x * 256 + threadIdx.x) >> 5;   // 0..B*64-1 exactly
  const int b    = wv >> 6;
  const int rem  = wv & 63;
  const int d0   = (rem >> 2) * 16;
  const int u0   = (rem & 3) * 16;
  const int m    = lane & 15;
  const int kb   = (lane >> 4) * 2;
  const float* x2b = x2 + (size_t)b * cL * cD;
  v8f acc = {};
  for (int k = 0; k < cL; k += 4) {
    v2f a, bb;
    a.x  = x2b[(size_t)(k + kb)     * cD + d0 + m];   // A[d][j] = x2[j][d]
    a.y  = x2b[(size_t)(k + kb + 1) * cD + d0 + m];
    bb.x = w1[(size_t)(k + kb)     * cU + u0 + m];
    bb.y = w1[(size_t)(k + kb + 1) * cU + u0 + m];
    acc = WMMA_F32(a, bb, acc);
  }
  const int rv = (lane >> 4) * 8;
  float* Tp = T + ((size_t)(b * cD + d0 + rv)) * cU + u0 + m;
#pragma unroll
  for (int v = 0; v < 8; ++v) Tp[(size_t)v * cU] = acc[v];
}

// -------- match1: async-LDS staged batched GEMM + cosine scaling --------
// Block: 8 waves, macro-tile 128(i) x 128(j); wave tile 32 x 64 (2x4 WMMA tiles).
// K pipelined in KC=64 chunks with GLOBAL_LOAD_ASYNC_TO_LDS_B128 (double buffer).
__global__ __launch_bounds__(256) void dot_match1_kernel(
    const float* __restrict__ x1, const float* __restrict__ x2,
    const float* __restrict__ rn1, const float* __restrict__ rn2,
    float* __restrict__ out1) {
  __shared__ float smem[2 * BUF_DW];          // ~136 KB of the 320 KB WGP LDS
  const uint32_t sbase = (uint32_t)(uintptr_t)&smem[0];

  const int t     = threadIdx.x;
  const int b     = blockIdx.z;
  const int iBase = blockIdx.y * 128;
  const int jBase = blockIdx.x * 128;
  const float* g1 = x1 + ((size_t)(b * cL + iBase)) * cD;
  const float* g2 = x2 + ((size_t)(b * cL + jBase)) * cD;

  // cooperative copy mapping: 16 lanes cover one 64-dword row slice (256B, coalesced)
  const int crow = t >> 4;
  const int ccol = (t & 15) * 4;

  auto issue_chunk = [&](int k0, int buf) {
    const uint32_t l1 = sbase + (uint32_t)(buf * BUF_DW) * 4u;
    const uint32_t l2 = l1 + (uint32_t)TILE_DW * 4u;
#pragma unroll
    for (int s = 0; s < 8; ++s) {
      const int r = s * 16 + crow;
      const uint32_t loff = (uint32_t)((r * LSTR + ccol) * 4);
      const float* ga = g1 + (size_t)r * cD + (k0 + ccol);
      const float* gb = g2 + (size_t)r * cD + (k0 + ccol);
      const uint32_t la = l1 + loff;
      const uint32_t lb = l2 + loff;
      asm volatile("global_load_async_to_lds_b128 %0, %1, off"
                   :: "v"(la), "v"(ga) : "memory");
      asm volatile("global_load_async_to_lds_b128 %0, %1, off"
                   :: "v"(lb), "v"(gb) : "memory");
    }
  };

  const int lane = t & 31;
  const int w    = t >> 5;
  const int wr   = w >> 1;                 // 0..3 -> i offset wr*32
  const int wc   = w & 1;                  // 0..1 -> j offset wc*64
  const int m    = lane & 15;
  const int kb   = (lane >> 4) * 2;
  const int arow = (wr * 32 + m) * LSTR + kb;
  const int brow = (wc * 64 + m) * LSTR + kb;

  v8f acc[2][4] = {};

  issue_chunk(0, 0);
  for (int c = 0; c < 4; ++c) {
    const int buf = c & 1;
    if (c < 3) {
      issue_chunk((c + 1) * KC, buf ^ 1);
      // 16 async ops just issued for chunk c+1; <=16 outstanding means chunk c done
      asm volatile("s_wait_asynccnt 0x10" ::: "memory");
    } else {
      asm volatile("s_wait_asynccnt 0x0" ::: "memory");
    }
    __syncthreads();

    const float* L1 = &smem[buf * BUF_DW];
    const float* L2 = L1 + TILE_DW;
#pragma unroll 2
    for (int k = 0; k < KC; k += 4) {
      v2f a0 = *(const v2f*)&L1[arow + k];
      v2f a1 = *(const v2f*)&L1[arow + 16 * LSTR + k];
      v2f b0 = *(const v2f*)&L2[brow + k];
      v2f b1 = *(const v2f*)&L2[brow + 16 * LSTR + k];
      v2f b2 = *(const v2f*)&L2[brow + 32 * LSTR + k];
      v2f b3 = *(const v2f*)&L2[brow + 48 * LSTR + k];
      acc[0][0] = WMMA_F32(a0, b0, acc[0][0]);
      acc[0][1] = WMMA_F32(a0, b1, acc[0][1]);
      acc[0][2] = WMMA_F32(a0, b2, acc[0][2]);
      acc[0][3] = WMMA_F32(a0, b3, acc[0][3]);
      acc[1][0] = WMMA_F32(a1, b0, acc[1][0]);
      acc[1][1] = WMMA_F32(a1, b1, acc[1][1]);
      acc[1][2] = WMMA_F32(a1, b2, acc[1][2]);
      acc[1][3] = WMMA_F32(a1, b3, acc[1][3]);
    }
    __syncthreads();
  }

  const int rv = (lane >> 4) * 8;
  const float* r1 = rn1 + b * cL;
  const float* r2 = rn2 + b * cL;
  float* outb = out1 + (size_t)b * cL * cL;
#pragma unroll
  for (int ti = 0; ti < 2; ++ti) {
#pragma unroll
    for (int tj = 0; tj < 4; ++tj) {
      const int r0 = iBase + wr * 32 + ti * 16 + rv;
      const int cj = jBase + wc * 64 + tj * 16 + m;
      const float s2 = r2[cj];
      const float* rp = r1 + r0;
      float* orow = outb + (size_t)r0 * cL + cj;
      v8f cc = acc[ti][tj];
#pragma unroll
      for (int v = 0; v < 8; ++v)
        orow[(size_t)v * cL] = cc[v] * rp[v] * s2;
    }
  }
}

// -------- match2[b] = x1[b] @ T[b] : (L x U), K = D --------
__global__ __launch_bounds__(256) void match2_kernel(
    const float* __restrict__ x1, const float* __restrict__ T,
    float* __restrict__ out2) {
  const int lane = threadIdx.x & 31;
  const int wv   = (blockIdx.x * 256 + threadIdx.x) >> 5;  // 0..B*256-1 exactly
  const int b    = wv >> 8;
  const int rem  = wv & 255;
  const int i0   = (rem >> 2) * 16;
  const int u0   = (rem & 3) * 16;
  const int m    = lane & 15;
  const int kb   = (lane >> 4) * 2;
  const float* Ap = x1 + ((size_t)(b * cL + i0 + m)) * cD + kb;
  const float* Tb = T + (size_t)b * cD * cU;
  v8f acc = {};
#pragma unroll 4
  for (int k = 0; k < cD; k += 4) {
    v2f a = *(const v2f*)(Ap + k);
    v2f bb;
    bb.x = Tb[(size_t)(k + kb)     * cU + u0 + m];
    bb.y = Tb[(size_t)(k + kb + 1) * cU + u0 + m];
    acc = WMMA_F32(a, bb, acc);
  }
  const int rv = (lane >> 4) * 8;
  float* op = out2 + ((size_t)(b * cL + i0 + rv)) * cU + u0 + m;
#pragma unroll
  for (int v = 0; v < 8; ++v) op[(size_t)v * cU] = acc[v];
}

extern "C" void kernel_launch(void* const* d_in, const int* in_sizes, int n_in,
                              void* d_out, int out_size, void* d_ws, size_t ws_size,
                              hipStream_t stream) {
  (void)in_sizes; (void)n_in; (void)out_size; (void)ws_size;
  const float* x1 = (const float*)d_in[0];
  const float* x2 = (const float*)d_in[1];
  const float* w1 = (const float*)d_in[2];

  float* out  = (float*)d_out;
  float* out1 = out;                                   // (B, L, L)
  float* out2 = out1 + (size_t)cB * cL * cL;           // (B, L, U)
  float* out3 = out2 + (size_t)cB * cL * cU;           // (B, L, 6D)

  float* ws  = (float*)d_ws;
  float* rn1 = ws;                    // B*L floats
  float* rn2 = ws + cB * cL;          // B*L floats
  float* T   = ws + 2 * cB * cL;      // B*D*U floats (~1 MB)

  match3_norms_kernel<<<(cB * cL) / 8, 256, 0, stream>>>(x1, x2, out3, rn1, rn2);
  xtw_kernel<<<(cB * 64) / 8, 256, 0, stream>>>(x2, w1, T);
  dot_match1_kernel<<<dim3(cL / 128, cL / 128, cB), 256, 0, stream>>>(
      x1, x2, rn1, rn2, out1);
  match2_kernel<<<(cB * 256) / 8, 256, 0, stream>>>(x1, T, out2);
}